// LightGCNAgg_37890201485520
// MI455X (gfx1250) — compile-verified
//
#include <hip/hip_runtime.h>
#include <stdint.h>

// LightGCN aggregation: COO SpMM  out[r,:] += v * x[c,:], DIM = 32 (== wave32).
// Roofline: 2*E*D = 102 MFLOP vs ~45 MB L2-resident traffic (x and out are
// 12.8 MB each, well inside the 192 MB L2) -> memory/atomic bound. WMMA gives
// nothing (random scatter, no dense tiles); the CDNA5 win is the async
// global->LDS edge-streaming path + wave-wide 128B coalesced f32 atomics.

#define DIM 32

constexpr int CHUNK    = 256;   // edges per staged chunk (256 * 4B per array)
constexpr int NTHREADS = 256;   // 8 waves; wave w consumes edges [w*32, w*32+32)
constexpr int EDGES_PER_WAVE = CHUNK / (NTHREADS / 32); // 32
constexpr int MAX_BLOCKS = 2500; // ~20k waves to keep L2 atomic pipes saturated

__global__ void zero_f32(float* __restrict__ p, int n) {
  int i4 = (blockIdx.x * blockDim.x + threadIdx.x) * 4;
  if (i4 + 3 < n) {
    float4 z; z.x = 0.f; z.y = 0.f; z.z = 0.f; z.w = 0.f;
    *(float4*)(p + i4) = z;
  } else if (i4 < n) {                      // <=3 scalar stores, last thread only
    p[i4] = 0.f;
    if (i4 + 1 < n) p[i4 + 1] = 0.f;
    if (i4 + 2 < n) p[i4 + 2] = 0.f;
  }
}

__device__ __forceinline__ void atomic_add_f32(float* addr, float v) {
  // relaxed, agent scope -> native no-return global_atomic_add_f32
  __hip_atomic_fetch_add(addr, v, __ATOMIC_RELAXED, __HIP_MEMORY_SCOPE_AGENT);
}

__global__ __launch_bounds__(NTHREADS) void spmm_coo_async(
    const int*   __restrict__ rows,
    const int*   __restrict__ cols,
    const float* __restrict__ vals,
    const float* __restrict__ x,
    float*       __restrict__ out,
    int nchunks)
{
  __shared__ int   s_rows[2][CHUNK];
  __shared__ int   s_cols[2][CHUNK];
  __shared__ float s_vals[2][CHUNK];

  const int t    = threadIdx.x;
  const int wid  = t >> 5;
  const int lane = t & 31;
  const unsigned voff = (unsigned)t * 4u;   // byte offset of this lane's element

  // Stage one chunk of edge data into LDS buffer `buf` via CDNA5 async copies.
  // Generic pointers to LDS are {SHARED_BASE_hi32, lds_offset_lo32}; the async
  // instruction takes the low-32 LDS byte offset in the vdst VGPR.
  auto stage = [&](int chunk, int buf) {
    const unsigned lr = (unsigned)(uintptr_t)&s_rows[buf][0] + voff;
    const unsigned lc = (unsigned)(uintptr_t)&s_cols[buf][0] + voff;
    const unsigned lv = (unsigned)(uintptr_t)&s_vals[buf][0] + voff;
    const int*   gr = rows + (size_t)chunk * CHUNK;
    const int*   gc = cols + (size_t)chunk * CHUNK;
    const float* gv = vals + (size_t)chunk * CHUNK;
    asm volatile("global_load_async_to_lds_b32 %0, %1, %2"
                 :: "v"(lr), "v"(voff), "s"(gr) : "memory");
    asm volatile("global_load_async_to_lds_b32 %0, %1, %2"
                 :: "v"(lc), "v"(voff), "s"(gc) : "memory");
    asm volatile("global_load_async_to_lds_b32 %0, %1, %2"
                 :: "v"(lv), "v"(voff), "s"(gv) : "memory");
  };

  const int stride = gridDim.x;
  const int c0 = blockIdx.x;
  if (c0 >= nchunks) return;

  stage(c0, 0);                         // prologue: fill buffer 0
  int buf = 0;

  for (int c = c0; c < nchunks; c += stride) {
    const int nxt = c + stride;         // uniform across the block
    if (nxt < nchunks) {
      stage(nxt, buf ^ 1);              // prefetch next chunk into other buffer
      // Async loads retire in order per wave: ASYNCcnt<=3 means the oldest 3
      // (this chunk) are done while the newest 3 (next chunk) stay in flight.
      asm volatile("s_wait_asynccnt 0x3" ::: "memory");
    } else {
      asm volatile("s_wait_asynccnt 0x0" ::: "memory");
    }
    __syncthreads();

    const int ebase = wid * EDGES_PER_WAVE;
    #pragma unroll 8
    for (int j = 0; j < EDGES_PER_WAVE; ++j) {
      const int   r  = s_rows[buf][ebase + j];   // uniform ds_load_b128 groups
      const int   cc = s_cols[buf][ebase + j];
      const float v  = s_vals[buf][ebase + j];
      const float xv = x[(size_t)cc * DIM + lane];   // coalesced 128B, L2-hot
      atomic_add_f32(&out[(size_t)r * DIM + lane], v * xv); // 1 L2 line / wave
    }
    __syncthreads();                    // readers done before buffer is reused
    buf ^= 1;
  }
}

// Tail edges (E not divisible by CHUNK): one edge per wave, no staging.
__global__ void spmm_coo_tail(
    const int* __restrict__ rows, const int* __restrict__ cols,
    const float* __restrict__ vals, const float* __restrict__ x,
    float* __restrict__ out, int ebase, int E)
{
  int gid  = blockIdx.x * blockDim.x + threadIdx.x;
  int e    = ebase + (gid >> 5);
  int lane = gid & 31;
  if (e < E) {
    const int   r  = rows[e];
    const int   cc = cols[e];
    const float v  = vals[e];
    const float xv = x[(size_t)cc * DIM + lane];
    atomic_add_f32(&out[(size_t)r * DIM + lane], v * xv);
  }
}

extern "C" void kernel_launch(void* const* d_in, const int* in_sizes, int n_in,
                              void* d_out, int out_size, void* d_ws, size_t ws_size,
                              hipStream_t stream) {
  const int*   rows = (const int*)  d_in[0];
  const int*   cols = (const int*)  d_in[1];
  const float* vals = (const float*)d_in[2];
  const float* x    = (const float*)d_in[3];
  float*       out  = (float*)      d_out;
  const int E = in_sizes[0];

  // 1) zero the (poisoned) output
  {
    int nthr = (out_size + 3) / 4;
    int nblk = (nthr + 255) / 256;
    zero_f32<<<nblk, 256, 0, stream>>>(out, out_size);
  }

  // 2) main async-staged atomic-scatter SpMM over full chunks
  const int nchunks = E / CHUNK;
  if (nchunks > 0) {
    // Need >=2 chunks/block for double buffering to overlap; cap grid so most
    // blocks run 2-3 chunks while keeping ~20k waves resident.
    int nblocks = nchunks < MAX_BLOCKS ? nchunks : MAX_BLOCKS;
    spmm_coo_async<<<nblocks, NTHREADS, 0, stream>>>(rows, cols, vals, x, out,
                                                     nchunks);
  }

  // 3) remainder edges
  const int ebase = nchunks * CHUNK;
  const int rem   = E - ebase;
  if (rem > 0) {
    int nthr = rem * 32;
    int nblk = (nthr + 255) / 256;
    spmm_coo_tail<<<nblk, 256, 0, stream>>>(rows, cols, vals, x, out, ebase, E);
  }
}